// DecomLayer_50611894616605
// MI455X (gfx1250) — compile-verified
//
#include <hip/hip_runtime.h>
#include <math.h>

// Problem constants (fixed by the reference)
#define B_    64
#define N_    2048
#define H_    64
#define NH_   4
#define DH_   16
#define M_    (3 * N_)      // 6144
#define E_    (8 * M_)      // 49152
#define SPLITS 8
#define NORM_ 0.25f         // 1/sqrt(DH)

typedef __attribute__((ext_vector_type(2))) float v2f;
typedef __attribute__((ext_vector_type(8))) float v8f;

// ---------------------------------------------------------------------------
// K0: zero the pooled accumulator (d_ws is not re-poisoned between replays)
// ---------------------------------------------------------------------------
__global__ void zero_pool_kernel(float* __restrict__ pool) {
    int i = blockIdx.x * blockDim.x + threadIdx.x;
    if (i < B_ * 3 * H_) pool[i] = 0.0f;
}

// ---------------------------------------------------------------------------
// K1: fused per-graph COO SpMM + segment-pool.
//   pool[b, d_index[b, rows[e]], :] += vals[e] * x[b, cols[e], :]
// Grid: (B, SPLITS) x 256 threads.
// Thread layout: 16 channel-groups (float4 = 4 channels) x 16 edge slices.
// Private 3x float4 accumulators -> LDS ds_add_f32 -> global atomic add.
// ---------------------------------------------------------------------------
__global__ __launch_bounds__(256) void spmm_pool_kernel(
    const float* __restrict__ x,      // [B*N, H]
    const int*   __restrict__ rows,   // [B, E]
    const int*   __restrict__ cols,   // [B, E]
    const float* __restrict__ vals,   // [B, E]
    const int*   __restrict__ dindex, // [B, M]
    float*       __restrict__ pool)   // [B, 3, H]
{
    const int b     = blockIdx.x;
    const int split = blockIdx.y;
    const int group = threadIdx.x & 15;   // channel group: channels [4g, 4g+3]
    const int slice = threadIdx.x >> 4;   // edge slice 0..15

    const int    chunk = E_ / SPLITS;
    const int    e0    = split * chunk;
    const int    ebase = b * E_;
    const float* xg    = x + b * (N_ * H_);
    const int*   idxg  = dindex + b * M_;

    float4 acc0 = make_float4(0.f, 0.f, 0.f, 0.f);
    float4 acc1 = acc0;
    float4 acc2 = acc0;

    for (int e = e0 + slice; e < e0 + chunk; e += 16) {
        const int   r   = rows[ebase + e];
        const int   c   = cols[ebase + e];
        const float v   = vals[ebase + e];
        const int   idx = idxg[r];

        const float4 xv =
            *reinterpret_cast<const float4*>(xg + c * H_ + group * 4);

        // branchless row select: 12 FMAs per edge per thread
        const float w0 = (idx == 0) ? v : 0.f;
        const float w1 = (idx == 1) ? v : 0.f;
        const float w2 = (idx == 2) ? v : 0.f;

        acc0.x = fmaf(w0, xv.x, acc0.x); acc0.y = fmaf(w0, xv.y, acc0.y);
        acc0.z = fmaf(w0, xv.z, acc0.z); acc0.w = fmaf(w0, xv.w, acc0.w);
        acc1.x = fmaf(w1, xv.x, acc1.x); acc1.y = fmaf(w1, xv.y, acc1.y);
        acc1.z = fmaf(w1, xv.z, acc1.z); acc1.w = fmaf(w1, xv.w, acc1.w);
        acc2.x = fmaf(w2, xv.x, acc2.x); acc2.y = fmaf(w2, xv.y, acc2.y);
        acc2.z = fmaf(w2, xv.z, acc2.z); acc2.w = fmaf(w2, xv.w, acc2.w);
    }

    __shared__ float ps[3 * H_];
    for (int i = threadIdx.x; i < 3 * H_; i += 256) ps[i] = 0.f;
    __syncthreads();

    const int cb = group * 4;
    atomicAdd(&ps[0 * H_ + cb + 0], acc0.x);
    atomicAdd(&ps[0 * H_ + cb + 1], acc0.y);
    atomicAdd(&ps[0 * H_ + cb + 2], acc0.z);
    atomicAdd(&ps[0 * H_ + cb + 3], acc0.w);
    atomicAdd(&ps[1 * H_ + cb + 0], acc1.x);
    atomicAdd(&ps[1 * H_ + cb + 1], acc1.y);
    atomicAdd(&ps[1 * H_ + cb + 2], acc1.z);
    atomicAdd(&ps[1 * H_ + cb + 3], acc1.w);
    atomicAdd(&ps[2 * H_ + cb + 0], acc2.x);
    atomicAdd(&ps[2 * H_ + cb + 1], acc2.y);
    atomicAdd(&ps[2 * H_ + cb + 2], acc2.z);
    atomicAdd(&ps[2 * H_ + cb + 3], acc2.w);
    __syncthreads();

    for (int i = threadIdx.x; i < 3 * H_; i += 256)
        atomicAdd(&pool[b * 3 * H_ + i], ps[i]);
}

// ---------------------------------------------------------------------------
// K2: Q/K/V projections via V_WMMA_F32_16X16X4_F32 (exact fp32).
//   out = pool [192 x 64] @ W^T, i.e. B[k][n] = W[n][k].
// Grid: (12 M-tiles, 3 matrices), 128 threads (4 waves, one 16-col N-tile each).
// Fragment maps per ISA 7.12.2 (f32):
//   A 16x4 : lane -> M = lane&15, half-wave selects K pair, slots = K, K+1
//   C/D    : VGPR r -> M = r + 8*(lane>>4), lane&15 -> N
//   B 4x16 : mirrored from A (lane&15 -> N, half-wave selects K pair)
// ---------------------------------------------------------------------------
__global__ __launch_bounds__(128) void qkv_wmma_kernel(
    const float* __restrict__ pool,  // [192, 64]
    const float* __restrict__ Wq,
    const float* __restrict__ Wk,
    const float* __restrict__ Wv,
    float* __restrict__ Q,           // [192, 64] each
    float* __restrict__ K,
    float* __restrict__ V)
{
    const int mt   = blockIdx.x;          // 0..11  (M tile of 16 rows)
    const int mat  = blockIdx.y;          // 0=Q 1=K 2=V
    const int nt   = threadIdx.x >> 5;    // wave id -> N tile 0..3
    const int lane = threadIdx.x & 31;

    const float* W   = (mat == 0) ? Wq : (mat == 1) ? Wk : Wv;
    float*       out = (mat == 0) ? Q  : (mat == 1) ? K  : V;

    const int mrow = lane & 15;
    const int ksel = (lane >> 4) << 1;        // 0 or 2: this half-wave's K pair
    const int ncol = nt * 16 + (lane & 15);

    v8f acc = {0.f, 0.f, 0.f, 0.f, 0.f, 0.f, 0.f, 0.f};

#pragma unroll
    for (int s = 0; s < 16; ++s) {            // 16 x (K=4) = K of 64
        const int k0 = s * 4 + ksel;

        const float* ap = pool + (mt * 16 + mrow) * H_ + k0;
        v2f a;
        a.x = ap[0];
        a.y = ap[1];

        const float* bp = W + ncol * H_ + k0; // B = W^T: contiguous in k
        v2f bf;
        bf.x = bp[0];
        bf.y = bp[1];

        acc = __builtin_amdgcn_wmma_f32_16x16x4_f32(
            false, a, false, bf, (short)0, acc, false, false);
    }

    const int rbase = (lane >> 4) * 8;
#pragma unroll
    for (int r = 0; r < 8; ++r)
        out[(mt * 16 + rbase + r) * H_ + ncol] = acc[r];
}

// ---------------------------------------------------------------------------
// K3: 3-token, 4-head softmax attention. One thread per (graph, head).
// out[b, q*64 + h*16 + d] = sum_j softmax(Q.K^T/4)[q][j] * V[j][d]
// ---------------------------------------------------------------------------
__global__ __launch_bounds__(256) void attention_kernel(
    const float* __restrict__ Q,
    const float* __restrict__ K,
    const float* __restrict__ V,
    float* __restrict__ out)   // [B, 192]
{
    const int t = threadIdx.x;     // 256 = 64 graphs * 4 heads
    const int b = t >> 2;
    const int h = t & 3;

    const float* qb = Q + (b * 3) * H_ + h * DH_;
    const float* kb = K + (b * 3) * H_ + h * DH_;
    const float* vb = V + (b * 3) * H_ + h * DH_;

    float p[3][3];
#pragma unroll
    for (int i = 0; i < 3; ++i) {
#pragma unroll
        for (int j = 0; j < 3; ++j) {
            float acc = 0.f;
#pragma unroll
            for (int d = 0; d < DH_; ++d)
                acc = fmaf(qb[i * H_ + d], kb[j * H_ + d], acc);
            p[i][j] = acc * NORM_;
        }
        // softmax over j
        float mx = fmaxf(p[i][0], fmaxf(p[i][1], p[i][2]));
        float e0 = __expf(p[i][0] - mx);
        float e1 = __expf(p[i][1] - mx);
        float e2 = __expf(p[i][2] - mx);
        float inv = 1.f / (e0 + e1 + e2);
        p[i][0] = e0 * inv; p[i][1] = e1 * inv; p[i][2] = e2 * inv;
    }

#pragma unroll
    for (int i = 0; i < 3; ++i) {
#pragma unroll
        for (int d = 0; d < DH_; ++d) {
            float o = p[i][0] * vb[0 * H_ + d];
            o = fmaf(p[i][1], vb[1 * H_ + d], o);
            o = fmaf(p[i][2], vb[2 * H_ + d], o);
            out[b * (3 * H_) + i * H_ + h * DH_ + d] = o;
        }
    }
}

// ---------------------------------------------------------------------------
extern "C" void kernel_launch(void* const* d_in, const int* in_sizes, int n_in,
                              void* d_out, int out_size, void* d_ws, size_t ws_size,
                              hipStream_t stream) {
    const float* x      = (const float*)d_in[0];
    // d_in[1] = batch ids (implied by fixed N), d_in[2] = batch_size (=64) : unused
    const int*   rows   = (const int*)  d_in[3];
    const int*   cols   = (const int*)  d_in[4];
    const float* vals   = (const float*)d_in[5];
    const int*   dindex = (const int*)  d_in[6];
    const float* Wq     = (const float*)d_in[7];
    const float* Wk     = (const float*)d_in[8];
    const float* Wv     = (const float*)d_in[9];

    float* ws   = (float*)d_ws;
    float* pool = ws;                       // 192*64 = 12288 floats
    float* Q    = ws + 12288;
    float* K    = ws + 24576;
    float* V    = ws + 36864;
    float* out  = (float*)d_out;            // [64, 192]

    zero_pool_kernel<<<(B_ * 3 * H_ + 255) / 256, 256, 0, stream>>>(pool);
    spmm_pool_kernel<<<dim3(B_, SPLITS), 256, 0, stream>>>(
        x, rows, cols, vals, dindex, pool);
    qkv_wmma_kernel<<<dim3(12, 3), 128, 0, stream>>>(pool, Wq, Wk, Wv, Q, K, V);
    attention_kernel<<<1, 256, 0, stream>>>(Q, K, V, out);
}